// EGATConv_71468255805606
// MI455X (gfx1250) — compile-verified
//
#include <hip/hip_runtime.h>
#include <hip/hip_bf16.h>
#include <math.h>

// EGATConv for MI455X (gfx1250, wave32).
// Memory-bound workload: fp32 end-to-end, native fp32 WMMA
// (V_WMMA_F32_16X16X4_F32) for the dense projections, and CDNA5 async
// global->LDS copies (GLOBAL_LOAD_ASYNC_TO_LDS_B128, ASYNCcnt) for tile
// staging so the VGPR file and LOADcnt slots stay free for the edge gathers.

typedef __attribute__((ext_vector_type(2))) float v2f;
typedef __attribute__((ext_vector_type(8))) float v8f;

#define NDIM 128   // DIN
#define EDIM 32    // DEIN
#define HFE  64    // H*FE == H*FN
#define NH   4     // heads
#define NEG_SLOPE 0.2f

#define A_PITCH 132   // node A-tile LDS row pitch (floats): 16B aligned, conflict-free
#define E_PITCH 36    // edge tile LDS row pitch (floats): 16B aligned, conflict-free

// ---- CDNA5 async copy helpers -------------------------------------------
// LDS aperture: generic shared address carries the LDS byte offset in [31:0].
__device__ __forceinline__ unsigned lds_addr_of(const void* p) {
    return (unsigned)(uintptr_t)p;
}
__device__ __forceinline__ void async_load_b128(unsigned lds_byte_addr, const void* gptr) {
    asm volatile("global_load_async_to_lds_b128 %0, %1, off"
                 :: "v"(lds_byte_addr), "v"((unsigned long long)(uintptr_t)gptr)
                 : "memory");
}
__device__ __forceinline__ void wait_async0() {
    asm volatile("s_wait_asynccnt 0x0" ::: "memory");
}

// float atomic max via the signed/unsigned int ordering trick
// (valid when the cell is initialized to -inf; works for mixed signs).
__device__ __forceinline__ void atomic_max_f32(float* addr, float v) {
    if (v >= 0.0f) {
        atomicMax((int*)addr, __float_as_int(v));
    } else {
        atomicMin((unsigned int*)addr, (unsigned int)__float_as_int(v));
    }
}

// ---------------------------------------------------------------------------
// K0: init h_out (lives in d_out, harness poisons it), node_max, denom
// ---------------------------------------------------------------------------
__global__ void __launch_bounds__(256) egat_init_kernel(
    float* __restrict__ h_out, float* __restrict__ node_max,
    float* __restrict__ denom, int n)
{
    int idx = blockIdx.x * blockDim.x + threadIdx.x;
    if (idx < n * HFE) h_out[idx] = 0.0f;
    if (idx < n * NH) {
        node_max[idx] = -__builtin_inff();
        denom[idx]    = 0.0f;
    }
}

// ---------------------------------------------------------------------------
// K1: node projections f_ni / f_nj / h_src = nfeats @ {W_ni, W_nj, W_node}
// One block = 16 node rows, 4 waves; wave w owns output cols [16w,16w+16).
// fp32 WMMA 16x16x4, K=128 (32 steps), three accumulator chains per wave.
// A tile staged with async b128 copies (ASYNCcnt), weights streamed from L2.
// ---------------------------------------------------------------------------
__global__ void __launch_bounds__(128) egat_node_proj_kernel(
    const float* __restrict__ nfeats,
    const float* __restrict__ W_ni,
    const float* __restrict__ W_nj,
    const float* __restrict__ W_node,
    float* __restrict__ f_ni,
    float* __restrict__ f_nj,
    float* __restrict__ h_src)
{
    __shared__ float As[16 * A_PITCH];
    const int r0  = blockIdx.x * 16;
    const int tid = threadIdx.x;

    // 16 rows x 128 floats = 512 x 16B chunks; 4 per thread, async to LDS
    const unsigned lds_base = lds_addr_of(&As[0]);
    #pragma unroll
    for (int i = 0; i < 4; ++i) {
        const int c   = tid + i * 128;   // chunk id
        const int row = c >> 5;          // 32 chunks per row
        const int cw  = c & 31;
        async_load_b128(lds_base + (unsigned)(row * A_PITCH * 4 + cw * 16),
                        nfeats + (long)(r0 + row) * NDIM + cw * 4);
    }
    wait_async0();
    __syncthreads();

    const int wave = tid >> 5;
    const int lane = tid & 31;
    const int half = lane >> 4;             // K sub-pair select (A layout 16x4 f32)
    const int l16  = lane & 15;
    const int n0   = wave * 16;

    v8f c0 = {}; v8f c1 = {}; v8f c2 = {};
    #pragma unroll 4
    for (int k = 0; k < NDIM; k += 4) {
        v2f a;
        a.x = As[l16 * A_PITCH + k + 2 * half + 0];
        a.y = As[l16 * A_PITCH + k + 2 * half + 1];
        const int bi = (k + 2 * half) * HFE + n0 + l16;   // W is [K][64] row-major
        v2f b0, b1, b2;
        b0.x = W_ni[bi];   b0.y = W_ni[bi + HFE];
        b1.x = W_nj[bi];   b1.y = W_nj[bi + HFE];
        b2.x = W_node[bi]; b2.y = W_node[bi + HFE];
        c0 = __builtin_amdgcn_wmma_f32_16x16x4_f32(false, a, false, b0, (short)0, c0, false, false);
        c1 = __builtin_amdgcn_wmma_f32_16x16x4_f32(false, a, false, b1, (short)0, c1, false, false);
        c2 = __builtin_amdgcn_wmma_f32_16x16x4_f32(false, a, false, b2, (short)0, c2, false, false);
    }

    // C layout: VGPR i, lanes 0-15 -> M=i, lanes 16-31 -> M=i+8; N = n0+l16
    #pragma unroll
    for (int i = 0; i < 8; ++i) {
        const int row = r0 + i + half * 8;
        const int col = n0 + l16;
        f_ni [row * HFE + col] = c0[i];
        f_nj [row * HFE + col] = c1[i];
        h_src[row * HFE + col] = c2[i];
    }
}

// ---------------------------------------------------------------------------
// K2: per-edge: f_fij (WMMA, K=32) + gather f_ni[src]+f_nj[dst]+bias,
// leaky-relu -> f_out (d_out), attention dot -> e_logit, atomic max by dst.
// One block = 16 edges; wave w = head h = w. efeats tile staged async.
// ---------------------------------------------------------------------------
__global__ void __launch_bounds__(128) egat_edge_logit_kernel(
    const float* __restrict__ efeats,
    const float* __restrict__ W_fij,
    const float* __restrict__ bias,
    const float* __restrict__ attn,     // flat [H*FE]
    const int*   __restrict__ src,
    const int*   __restrict__ dst,
    const float* __restrict__ f_ni,
    const float* __restrict__ f_nj,
    float* __restrict__ f_out,          // [E][64] (second region of d_out)
    float* __restrict__ e_logit,        // [E][H]
    float* __restrict__ node_max)       // [N][H]
{
    __shared__ float Es[16 * E_PITCH];
    __shared__ int   s_src[16], s_dst[16];
    const int e0  = blockIdx.x * 16;
    const int tid = threadIdx.x;

    // 16 rows x 32 floats = 128 x 16B chunks; one per thread, async to LDS
    {
        const int row = tid >> 3;        // 8 chunks per row
        const int cw  = tid & 7;
        async_load_b128(lds_addr_of(&Es[0]) + (unsigned)(row * E_PITCH * 4 + cw * 16),
                        efeats + (long)(e0 + row) * EDIM + cw * 4);
    }
    if (tid < 16)      s_src[tid]      = src[e0 + tid];
    else if (tid < 32) s_dst[tid - 16] = dst[e0 + tid - 16];
    wait_async0();
    __syncthreads();

    const int wave = tid >> 5;          // == head h
    const int lane = tid & 31;
    const int half = lane >> 4;
    const int l16  = lane & 15;
    const int n0   = wave * 16;

    v8f c = {};
    #pragma unroll
    for (int k = 0; k < EDIM; k += 4) {
        v2f a, b;
        a.x = Es[l16 * E_PITCH + k + 2 * half + 0];
        a.y = Es[l16 * E_PITCH + k + 2 * half + 1];
        const int bi = (k + 2 * half) * HFE + n0 + l16;
        b.x = W_fij[bi]; b.y = W_fij[bi + HFE];
        c = __builtin_amdgcn_wmma_f32_16x16x4_f32(false, a, false, b, (short)0, c, false, false);
    }

    const int   col  = n0 + l16;
    const float bi_v = bias[col];
    const float at_v = attn[col];       // attn flat index h*FE+fe == col

    #pragma unroll
    for (int i = 0; i < 8; ++i) {
        const int r = i + half * 8;     // local edge in tile
        const int e = e0 + r;
        const int s = s_src[r];
        const int d = s_dst[r];
        float v = c[i] + f_ni[s * HFE + col] + f_nj[d * HFE + col] + bi_v;
        float f = (v > 0.0f) ? v : NEG_SLOPE * v;
        f_out[(long)e * HFE + col] = f;
        float p = f * at_v;
        // sum across the 16 lanes of this half (one head worth of FE)
        p += __shfl_xor(p, 1, 32);
        p += __shfl_xor(p, 2, 32);
        p += __shfl_xor(p, 4, 32);
        p += __shfl_xor(p, 8, 32);
        if (l16 == 0) {
            e_logit[e * NH + wave] = p;
            atomic_max_f32(&node_max[d * NH + wave], p);
        }
    }
}

// ---------------------------------------------------------------------------
// K3: p = exp(e - m[dst]); denom[dst] += p  (segment-softmax numerator/denom)
// ---------------------------------------------------------------------------
__global__ void __launch_bounds__(256) egat_softmax_kernel(
    const int*   __restrict__ dst,
    const float* __restrict__ e_logit,
    const float* __restrict__ node_max,
    float* __restrict__ p_buf,
    float* __restrict__ denom,
    long total)                          // E*H
{
    long idx = (long)blockIdx.x * blockDim.x + threadIdx.x;
    if (idx >= total) return;
    const int e = (int)(idx >> 2);
    const int h = (int)(idx & 3);
    const int d = dst[e];
    const float p = expf(e_logit[idx] - node_max[d * NH + h]);
    p_buf[idx] = p;
    atomicAdd(&denom[d * NH + h], p);
}

// ---------------------------------------------------------------------------
// K4: h_out[dst] += h_src[src] * (p / denom[dst])  (weighted aggregation)
// ---------------------------------------------------------------------------
__global__ void __launch_bounds__(256) egat_aggregate_kernel(
    const int*   __restrict__ src,
    const int*   __restrict__ dst,
    const float* __restrict__ p_buf,
    const float* __restrict__ denom,
    const float* __restrict__ h_src,
    float* __restrict__ h_out,
    long total)                          // E*64
{
    long idx = (long)blockIdx.x * blockDim.x + threadIdx.x;
    if (idx >= total) return;
    const int e   = (int)(idx >> 6);
    const int col = (int)(idx & 63);
    const int h   = col >> 4;
    const int d   = dst[e];
    const float a = p_buf[e * NH + h] / denom[d * NH + h];
    atomicAdd(&h_out[d * HFE + col], h_src[src[e] * HFE + col] * a);
}

// ---------------------------------------------------------------------------
extern "C" void kernel_launch(void* const* d_in, const int* in_sizes, int n_in,
                              void* d_out, int out_size, void* d_ws, size_t ws_size,
                              hipStream_t stream) {
    // setup_inputs order:
    const float* nfeats = (const float*)d_in[0];   // [N,128]
    const float* efeats = (const float*)d_in[1];   // [E,32]
    const float* W_ni   = (const float*)d_in[2];   // [128,64]
    const float* W_nj   = (const float*)d_in[3];   // [128,64]
    const float* W_fij  = (const float*)d_in[4];   // [32,64]
    const float* W_node = (const float*)d_in[5];   // [128,64]
    const float* attn   = (const float*)d_in[6];   // [1,4,16]
    const float* bias   = (const float*)d_in[7];   // [64]
    const int*   src    = (const int*)d_in[8];     // [E]
    const int*   dst    = (const int*)d_in[9];     // [E]

    const int N = in_sizes[0] / NDIM;   // 50000 (multiple of 16)
    const int E = in_sizes[1] / EDIM;   // 800000 (multiple of 16)

    // d_out = h_out [N*64] ++ f_out [E*64]
    float* h_out = (float*)d_out;
    float* f_out = h_out + (long)N * HFE;

    // workspace carve (floats): 3*N*64 + 2*E*4 + 2*N*4  ~= 66 MB
    float* ws       = (float*)d_ws;
    float* f_ni     = ws;
    float* f_nj     = f_ni  + (long)N * HFE;
    float* h_src    = f_nj  + (long)N * HFE;
    float* e_logit  = h_src + (long)N * HFE;
    float* p_buf    = e_logit + (long)E * NH;
    float* node_max = p_buf   + (long)E * NH;
    float* denom    = node_max + (long)N * NH;

    // K0: init
    {
        int total = N * HFE;
        egat_init_kernel<<<(total + 255) / 256, 256, 0, stream>>>(h_out, node_max, denom, N);
    }
    // K1: node projections (WMMA f32, async LDS staging)
    egat_node_proj_kernel<<<N / 16, 128, 0, stream>>>(
        nfeats, W_ni, W_nj, W_node, f_ni, f_nj, h_src);
    // K2: edge projection + gather + leakyrelu + logits + segment max
    egat_edge_logit_kernel<<<E / 16, 128, 0, stream>>>(
        efeats, W_fij, bias, attn, src, dst, f_ni, f_nj, f_out, e_logit, node_max);
    // K3: softmax numerators + denominators
    {
        long total = (long)E * NH;
        egat_softmax_kernel<<<(int)((total + 255) / 256), 256, 0, stream>>>(
            dst, e_logit, node_max, p_buf, denom, total);
    }
    // K4: weighted aggregation
    {
        long total = (long)E * HFE;
        egat_aggregate_kernel<<<(int)((total + 255) / 256), 256, 0, stream>>>(
            src, dst, p_buf, denom, h_src, h_out, total);
    }
}